// ADDMean_16595753632499
// MI455X (gfx1250) — compile-verified
//
#include <hip/hip_runtime.h>
#include <hip/hip_bf16.h>

// ---------------------------------------------------------------------------
// SPD Karcher-mean flow, all matrix functions (sqrtm/invsqrtm/logm/expm)
// recast as GEMM chains so everything lands on V_WMMA_F32_16X16X4_F32.
// n = 256, batch = 256. Working set fits the 192MB L2 -> compute bound.
// Staging: GLOBAL_LOAD_ASYNC_TO_LDS (ASYNCcnt) when the toolchain exposes it.
// ---------------------------------------------------------------------------

typedef __attribute__((ext_vector_type(2))) float v2f;
typedef __attribute__((ext_vector_type(8))) float v8f;
typedef int v4i_vs __attribute__((vector_size(16)));   // matches builtin param

#define NN   256          // matrix dim
#define N2   65536        // elements per matrix
#define TM   128
#define TN   128
#define TK   16

#define NS_ITERS 12       // Newton-Schulz iterations for sqrt/invsqrt
#define LOG_J    4        // square roots before log series
#define LOG_P    8        // Mercator series terms
#define EXP_S    5        // expm scaling (2^5)
#define EXP_P    8        // expm Taylor terms
#define K_OUTER  5        // reference K_ITERS

#define AS1 __attribute__((address_space(1)))
#define AS3 __attribute__((address_space(3)))

#if __has_builtin(__builtin_amdgcn_global_load_async_to_lds_b128) && \
    __has_builtin(__builtin_amdgcn_global_load_async_to_lds_b32)
#define USE_ASYNC_LDS 1
#else
#define USE_ASYNC_LDS 0
#endif

__device__ __forceinline__ unsigned long long lds_off32(const void* p) {
    // generic pointer to shared memory: low 32 bits = wave-relative LDS offset
    return (unsigned long long)(unsigned)(unsigned long long)p;
}

__device__ __forceinline__ void async_wait0() {
#if __has_builtin(__builtin_amdgcn_s_wait_asynccnt)
    __builtin_amdgcn_s_wait_asynccnt(0);
#else
    asm volatile("s_wait_asynccnt 0x0" ::: "memory");
#endif
}

// ---------------------------------------------------------------------------
// Batched GEMM:  C[b] = alpha * A[b] x B[b] + beta * C[b] + gammaI * I
// strides in elements; stride 0 broadcasts a single matrix across the batch.
// 256 threads = 8 waves; 128x128 tile per block; 32x64 tile per wave.
// A tile LDS: row-major [128][20]  (fragment = contiguous K pair)
// B tile LDS: N-major  [128][18]  (fragment = contiguous K pair, no repack)
// ---------------------------------------------------------------------------
__global__ __launch_bounds__(256)
void k_gemm(const float* __restrict__ A, long sA,
            const float* __restrict__ B, long sB,
            float* __restrict__ C, long sC,
            float alpha, float beta, float gammaI)
{
    __shared__ float As[TM][20];    // row stride 20: 16B-aligned, conflict-free
    __shared__ float Bs[TN][18];    // N-major; stride 18: conflict-free gathers

    const int b  = blockIdx.z;
    const int m0 = blockIdx.y * TM;
    const int n0 = blockIdx.x * TN;
    const float* __restrict__ Ab = A + (long)b * sA;
    const float* __restrict__ Bb = B + (long)b * sB;
    float* __restrict__ Cb       = C + (long)b * sC;

    const int tid  = threadIdx.x;
    const int lane = tid & 31;
    const int wave = tid >> 5;
    const int half = lane >> 4;     // 0: lanes 0-15, 1: lanes 16-31
    const int l16  = lane & 15;
    const int wm0  = (wave & 3) * 32;    // 4 waves down M
    const int wn0  = (wave >> 2) * 64;   // 2 waves across N

    v8f zero = {};
    v8f acc[2][4];
    #pragma unroll
    for (int mi = 0; mi < 2; ++mi)
        #pragma unroll
        for (int ni = 0; ni < 4; ++ni) acc[mi][ni] = zero;

    for (int k0 = 0; k0 < NN; k0 += TK) {
        // ---- stage K-slice into LDS ----
#if USE_ASYNC_LDS
        #pragma unroll
        for (int i = 0; i < 2; ++i) {               // A: 2 x b128 per thread
            int idx = tid + i * 256;
            int ra  = idx >> 2;                     // 0..127
            int ca  = (idx & 3) << 2;               // 0,4,8,12
            AS1 v4i_vs* g = (AS1 v4i_vs*)(unsigned long long)
                            (Ab + (long)(m0 + ra) * NN + k0 + ca);
            AS3 v4i_vs* l = (AS3 v4i_vs*)lds_off32(&As[ra][ca]);
            __builtin_amdgcn_global_load_async_to_lds_b128(g, l, 0, 0);
        }
        #pragma unroll
        for (int i = 0; i < 8; ++i) {               // B: 8 x b32, transposed
            int idx = tid + i * 256;
            int kb  = idx >> 7;                     // 0..15
            int nb  = idx & 127;                    // 0..127
            AS1 int* g = (AS1 int*)(unsigned long long)
                         (Bb + (long)(k0 + kb) * NN + n0 + nb);
            AS3 int* l = (AS3 int*)lds_off32(&Bs[nb][kb]);
            __builtin_amdgcn_global_load_async_to_lds_b32(g, l, 0, 0);
        }
        async_wait0();
#else
        #pragma unroll
        for (int i = 0; i < 2; ++i) {
            int idx = tid + i * 256;
            int ra  = idx >> 2;
            int ca  = (idx & 3) << 2;
            float4 va = *(const float4*)(Ab + (long)(m0 + ra) * NN + k0 + ca);
            *(float4*)(&As[ra][ca]) = va;
        }
        #pragma unroll
        for (int i = 0; i < 8; ++i) {
            int idx = tid + i * 256;
            int kb  = idx >> 7;
            int nb  = idx & 127;
            Bs[nb][kb] = Bb[(long)(k0 + kb) * NN + n0 + nb];
        }
#endif
        if (k0 + TK < NN) {                         // warm L2 for next slice
            __builtin_prefetch(Ab + (long)(m0 + (tid >> 1)) * NN + k0 + TK, 0, 1);
            __builtin_prefetch(Bb + (long)(k0 + (tid >> 4)) * NN + n0 + (tid & 15) * 8, 0, 1);
        }
        __syncthreads();

        // ---- 4 x (2x4) WMMA f32 16x16x4 per K-slice ----
        #pragma unroll
        for (int kk = 0; kk < TK; kk += 4) {
            v2f af[2], bf[4];
            #pragma unroll
            for (int mi = 0; mi < 2; ++mi) {
                // ISA 7.12.2: 32-bit A 16x4 -> VGPR0: K={0|2}, VGPR1: K={1|3}
                int r = wm0 + mi * 16 + l16;
                af[mi] = *(const v2f*)&As[r][kk + 2 * half];
            }
            #pragma unroll
            for (int ni = 0; ni < 4; ++ni) {
                int c = wn0 + ni * 16 + l16;
                bf[ni] = *(const v2f*)&Bs[c][kk + 2 * half];
            }
            #pragma unroll
            for (int mi = 0; mi < 2; ++mi)
                #pragma unroll
                for (int ni = 0; ni < 4; ++ni)
                    acc[mi][ni] = __builtin_amdgcn_wmma_f32_16x16x4_f32(
                        false, af[mi], false, bf[ni],
                        (short)0, acc[mi][ni], false, false);
        }
        __syncthreads();
    }

    // ---- epilogue: alpha/beta + fused gammaI on the diagonal ----
    #pragma unroll
    for (int mi = 0; mi < 2; ++mi)
        #pragma unroll
        for (int ni = 0; ni < 4; ++ni)
            #pragma unroll
            for (int r = 0; r < 8; ++r) {
                int gm = m0 + wm0 + mi * 16 + r + 8 * half;
                int gn = n0 + wn0 + ni * 16 + l16;
                float v = alpha * acc[mi][ni][r];
                if (beta != 0.f) v += beta * Cb[(long)gm * NN + gn];
                if (gm == gn)    v += gammaI;
                Cb[(long)gm * NN + gn] = v;
            }
}

// ---------------------------------------------------------------------------
// Elementwise helpers
// ---------------------------------------------------------------------------

// D[b] = a * A[b] * s[b]^expo + g * I      (A or s may be null)
__global__ __launch_bounds__(256)
void k_scale_pow(const float* __restrict__ A, float* __restrict__ D,
                 const float* __restrict__ s, float expo, float a, float g)
{
    unsigned idx = blockIdx.x * 256u + threadIdx.x;
    unsigned b = idx >> 16, e = idx & 0xFFFFu;
    float f = 1.f;
    if (s) f = powf(s[b], expo);
    float v = (A ? A[idx] : 0.f) * a * f;
    if ((e >> 8) == (e & 255u)) v += g;
    D[idx] = v;
}

// c[b] = max(trace(A[b]), eps)
__global__ __launch_bounds__(256)
void k_trace(const float* __restrict__ A, float* __restrict__ c)
{
    __shared__ float red[256];
    int b = blockIdx.x;
    red[threadIdx.x] = A[(long)b * N2 + threadIdx.x * (NN + 1)];
    __syncthreads();
    for (int s = 128; s > 0; s >>= 1) {
        if ((int)threadIdx.x < s) red[threadIdx.x] += red[threadIdx.x + s];
        __syncthreads();
    }
    if (threadIdx.x == 0) c[b] = fmaxf(red[0], 1e-12f);
}

// D[e] = mean over batch of A[b][e]
__global__ __launch_bounds__(256)
void k_mean(const float* __restrict__ A, float* __restrict__ D, int batch)
{
    unsigned e = blockIdx.x * 256u + threadIdx.x;
    float acc = 0.f;
    for (int b = 0; b < batch; ++b) acc += A[(long)b * N2 + e];
    D[e] = acc / (float)batch;
}

// w1[b] = softmax(weights[b])[1]
__global__ __launch_bounds__(256)
void k_softmax_w1(const float* __restrict__ w, float* __restrict__ w1, int B)
{
    int b = blockIdx.x * 256 + threadIdx.x;
    if (b < B) {
        float a = w[2 * b], c = w[2 * b + 1];
        float m = fmaxf(a, c);
        float e0 = __expf(a - m), e1 = __expf(c - m);
        w1[b] = e1 / (e0 + e1);
    }
}

// ---------------------------------------------------------------------------
// Host-side orchestration (all stream-ordered; graph-capture safe)
// ---------------------------------------------------------------------------
namespace {

struct Ctx {
    hipStream_t st;
    float *Y, *Z, *T, *Y2, *Z2;   // Newton-Schulz scratch (batched)
    float *c;                     // per-batch trace scalars
};

inline void swapp(float*& a, float*& b) { float* t = a; a = b; b = t; }

inline void gemm(hipStream_t st, const float* A, long sA, const float* B, long sB,
                 float* C, long sC, int batch, float alpha, float beta, float gI)
{
    dim3 g(TN == 128 ? 2 : 1, TM == 128 ? 2 : 1, batch);
    k_gemm<<<g, dim3(256), 0, st>>>(A, sA, B, sB, C, sC, alpha, beta, gI);
}

inline void scale_pow(hipStream_t st, const float* A, float* D, const float* s,
                      float expo, float a, float g, int batch)
{
    k_scale_pow<<<dim3(batch * (N2 / 256)), dim3(256), 0, st>>>(A, D, s, expo, a, g);
}

// sqrt/invsqrt via trace-normalized coupled Newton-Schulz:
//   Y0 = A/tr(A), Z0 = I;  T = 1.5I - 0.5 Z Y;  Y <- Y T;  Z <- T Z
//   sqrt(A) = sqrt(tr) * Y,  invsqrt(A) = Z / sqrt(tr)
void ns_sqrt(Ctx& x, const float* Ain, int batch, float* sqrtOut, float* invOut)
{
    const long S = N2;
    k_trace<<<dim3(batch), dim3(256), 0, x.st>>>(Ain, x.c);
    scale_pow(x.st, Ain, x.Y, x.c, -1.f, 1.f, 0.f, batch);        // Y = A/c
    scale_pow(x.st, nullptr, x.Z, nullptr, 0.f, 0.f, 1.f, batch); // Z = I
    float *Y = x.Y, *Z = x.Z, *Y2 = x.Y2, *Z2 = x.Z2;
    for (int it = 0; it < NS_ITERS; ++it) {
        gemm(x.st, Z, S, Y, S, x.T, S, batch, -0.5f, 0.f, 1.5f); // T = 1.5I-0.5ZY
        gemm(x.st, Y, S, x.T, S, Y2, S, batch, 1.f, 0.f, 0.f);   // Y <- Y T
        gemm(x.st, x.T, S, Z, S, Z2, S, batch, 1.f, 0.f, 0.f);   // Z <- T Z
        swapp(Y, Y2); swapp(Z, Z2);
    }
    if (sqrtOut) scale_pow(x.st, Y, sqrtOut, x.c,  0.5f, 1.f, 0.f, batch);
    if (invOut)  scale_pow(x.st, Z, invOut,  x.c, -0.5f, 1.f, 0.f, batch);
}

// logm via inverse scaling-and-squaring:
//   R = A^(1/2^J) (J Newton-Schulz sqrts), E = R - I,
//   log(A) ~= 2^J * E * Horner(sum (-1)^{p+1} E^{p-1} / p)
void logm(Ctx& x, const float* Sin, float* Lout, int batch,
          float* R, float* E, float* H, float* H2)
{
    const long S = N2;
    ns_sqrt(x, Sin, batch, R, nullptr);
    for (int j = 1; j < LOG_J; ++j) ns_sqrt(x, R, batch, R, nullptr);
    scale_pow(x.st, R, E, nullptr, 0.f, 1.f, -1.f, batch);        // E = R - I
    float cP = ((LOG_P & 1) ? 1.f : -1.f) / (float)LOG_P;
    scale_pow(x.st, nullptr, H, nullptr, 0.f, 0.f, cP, batch);    // H = cP * I
    float *h = H, *h2 = H2;
    for (int p = LOG_P - 1; p >= 1; --p) {
        float cp = ((p & 1) ? 1.f : -1.f) / (float)p;
        gemm(x.st, E, S, h, S, h2, S, batch, 1.f, 0.f, cp);       // h = cp I + E h
        swapp(h, h2);
    }
    gemm(x.st, E, S, h, S, Lout, S, batch, (float)(1 << LOG_J), 0.f, 0.f);
}

// expm via scaling-and-squaring with Taylor-Horner
void expm(Ctx& x, const float* Ain, float* Eout, int batch,
          float* Bsc, float* H, float* H2)
{
    const long S = N2;
    scale_pow(x.st, Ain, Bsc, nullptr, 0.f, 1.f / (float)(1 << EXP_S), 0.f, batch);
    scale_pow(x.st, nullptr, H, nullptr, 0.f, 0.f, 1.f, batch);   // H = I
    float *h = H, *h2 = H2;
    for (int p = EXP_P; p >= 1; --p) {
        gemm(x.st, Bsc, S, h, S, h2, S, batch, 1.f / (float)p, 0.f, 1.f);
        swapp(h, h2);
    }
    for (int t = 0; t < EXP_S; ++t) {                             // repeated squaring
        float* dst = (t == EXP_S - 1) ? Eout : h2;
        gemm(x.st, h, S, h, S, dst, S, batch, 1.f, 0.f, 0.f);
        if (t < EXP_S - 1) swapp(h, h2);
    }
}

} // namespace

extern "C" void kernel_launch(void* const* d_in, const int* in_sizes, int n_in,
                              void* d_out, int out_size, void* d_ws, size_t ws_size,
                              hipStream_t stream)
{
    const float* X = (const float*)d_in[0];      // (B,1,256,256) f32
    const float* W = (const float*)d_in[1];      // (B,2) f32
    float* out     = (float*)d_out;              // (B,1,256,256) f32
    const int B    = in_sizes[0] >> 16;          // elements / 65536
    const long S   = N2;

    // ---- workspace carve-up (floats) ----
    float* ws = (float*)d_ws;
    float* cbuf  = ws;            // 256
    float* w1    = ws + 512;      // 256
    float* SM    = ws + 1024;     // small single-matrix region: 8 x N2
    float* M     = SM + 0 * N2;
    float* Ms    = SM + 1 * N2;
    float* Mis   = SM + 2 * N2;
    float* Lmean = SM + 3 * N2;
    float* EM    = SM + 4 * N2;
    float* big   = ws + 1024 + 8 * (long)N2;
    auto BUF = [&](int i) { return big + (long)i * B * (long)N2; };
    float* Ybuf = BUF(0);
    float* Zbuf = BUF(1);
    float* Tbuf = BUF(2);
    float* Y2b  = BUF(3);
    float* Z2b  = BUF(4);
    float* Rb   = BUF(5);
    float* Eb   = BUF(6);   // also expm Bscaled
    float* Hb   = BUF(7);
    float* H2b  = BUF(8);
    float* Sb   = BUF(9);
    float* T1b  = BUF(10);
    float* Lb   = BUF(11);
    float* Fs   = BUF(12);
    float* Fis  = BUF(13);

    Ctx x{stream, Ybuf, Zbuf, Tbuf, Y2b, Z2b, cbuf};

    // w = softmax(weights);  M = mean_b X_b
    k_softmax_w1<<<dim3((B + 255) / 256), dim3(256), 0, stream>>>(W, w1, B);
    k_mean<<<dim3(N2 / 256), dim3(256), 0, stream>>>(X, M, B);

    // ---- Karcher flow ----
    for (int k = 0; k < K_OUTER; ++k) {
        ns_sqrt(x, M, 1, Ms, Mis);                            // Ms, Mis
        gemm(stream, X, S, Mis, 0, T1b, S, B, 1.f, 0.f, 0.f); // T1 = X Mis
        gemm(stream, Mis, 0, T1b, S, Sb, S, B, 1.f, 0.f, 0.f);// S  = Mis X Mis
        logm(x, Sb, Lb, B, Rb, Eb, Hb, H2b);                  // L_b = logm(S_b)
        k_mean<<<dim3(N2 / 256), dim3(256), 0, stream>>>(Lb, Lmean, B);
        expm(x, Lmean, EM, 1, Eb, Hb, H2b);                   // EM = expm(mean L)
        gemm(stream, EM, S, Ms, S, T1b, S, 1, 1.f, 0.f, 0.f);
        gemm(stream, Ms, S, T1b, S, M, S, 1, 1.f, 0.f, 0.f);  // M = Ms EM Ms
    }

    // ---- geodesic step per matrix ----
    ns_sqrt(x, X, B, Fs, Fis);                                // Fs, Fis
    gemm(stream, Fis, S, M, 0, T1b, S, B, 1.f, 0.f, 0.f);     // T1 = Fis M
    gemm(stream, T1b, S, Fis, S, Sb, S, B, 1.f, 0.f, 0.f);    // S  = Fis M Fis
    logm(x, Sb, Lb, B, Rb, Eb, Hb, H2b);                      // G  = logm(S)
    scale_pow(stream, Lb, Rb, w1, 1.f, 1.f, 0.f, B);          // R = w1_b * G
    expm(x, Rb, Sb, B, Eb, Hb, H2b);                          // S = expm(w1 G)
    gemm(stream, Fs, S, Sb, S, T1b, S, B, 1.f, 0.f, 0.f);     // T1 = Fs S
    gemm(stream, T1b, S, Fs, S, out, S, B, 1.f, 0.f, 0.f);    // out = Fs S Fs
}